// ContrastiveLoss_47519518163172
// MI455X (gfx1250) — compile-verified
//
#include <hip/hip_runtime.h>
#include <hip/hip_bf16.h>
#include <stdint.h>

// ---- problem constants (from reference) ----
#define BATCH   16
#define NROW    1048576          // 4*512*512 flattened per batch row
#define KSEL    1048             // int(0.001 * NROW)
#define BINS    2048
#define BLOCK   256
#define ROWBLK  256              // blocks per row: 256 * (256 thr * 16 elem) = 1,048,576
#define TIECAP  1024

// ---- workspace layout (uint32 word offsets) ----
#define OFF_H1     0             // 16 * 2048 level-1 histogram
#define OFF_H2     32768         // 16 * 2048 level-2 histogram
#define OFF_THR    65536         // 16 * uint4 {b1, r1, b2, r2}
#define OFF_TCNT   65600         // 16 tie counters
#define OFF_RSUM   65616         // 16 per-row float sums
#define ZERO_WORDS 65632         // everything above zeroed each launch
#define OFF_TIDX   65664         // 16 * TIECAP tie indices
#define OFF_TVAL   82048         // 16 * TIECAP tie out-values

typedef __attribute__((ext_vector_type(2))) float v2f;
typedef __attribute__((ext_vector_type(8))) float v8f;

// elementwise map shared by every pass (identical codegen -> identical bits)
__device__ __forceinline__ void map_elem(float p, float n, float& out, uint32_t& bits) {
    float d = n - p;
    out = __expf(0.1f * d);
    float t = out - 1.0f;
    bits = __float_as_uint(t * t);   // dist >= 0 -> bits monotone in dist
}

// Cross-lane wave sum via V_WMMA_F32_16X16X4_F32 with B = ones.
// A (16x4 f32) = 2 VGPRs/lane: lanes 0-15 feed K=0/1, lanes 16-31 feed K=2/3.
// With a = (x, 0) per lane and B all-ones, D[m,n] = x[lane m] + x[lane m+16].
// Each lane then sums its 8 D VGPRs (covers 8 of the 16 M-rows); one
// shfl_xor(16) add completes the 32-lane total in every lane.
__device__ __forceinline__ float wave_sum_wmma(float x) {
    v2f a; a[0] = x;    a[1] = 0.0f;
    v2f b; b[0] = 1.0f; b[1] = 1.0f;
    v8f c = {};
    c = __builtin_amdgcn_wmma_f32_16x16x4_f32(false, a, false, b,
                                              (short)0, c, false, false);
    float s = c[0] + c[1] + c[2] + c[3] + c[4] + c[5] + c[6] + c[7];
    s += __shfl_xor(s, 16, 32);
    return s;
}

// ---------------- kernels ----------------

__global__ void k_init(uint32_t* ws) {
    int i = blockIdx.x * blockDim.x + threadIdx.x;
    if (i < ZERO_WORDS) ws[i] = 0u;
}

__global__ void k_hist1(const float* __restrict__ pos,
                        const float* __restrict__ neg,
                        uint32_t* __restrict__ ws) {
    __shared__ uint32_t lh[BINS];
    for (int b = threadIdx.x; b < BINS; b += BLOCK) lh[b] = 0u;
    __syncthreads();

    const int row = blockIdx.x / ROWBLK;
    const int bir = blockIdx.x % ROWBLK;
    const float4* p4 = (const float4*)(pos + (size_t)row * NROW);
    const float4* n4 = (const float4*)(neg + (size_t)row * NROW);
    const int base = bir * 1024 + threadIdx.x;      // float4 index within row

    #pragma unroll
    for (int i = 0; i < 4; ++i) {
        int f = base + i * BLOCK;
        float4 P = p4[f];
        float4 Q = n4[f];
        float o; uint32_t bi;
        map_elem(P.x, Q.x, o, bi); atomicAdd(&lh[bi >> 21], 1u);
        map_elem(P.y, Q.y, o, bi); atomicAdd(&lh[bi >> 21], 1u);
        map_elem(P.z, Q.z, o, bi); atomicAdd(&lh[bi >> 21], 1u);
        map_elem(P.w, Q.w, o, bi); atomicAdd(&lh[bi >> 21], 1u);
    }
    __syncthreads();

    uint32_t* gh = ws + OFF_H1 + row * BINS;
    for (int b = threadIdx.x; b < BINS; b += BLOCK)
        if (lh[b]) atomicAdd(&gh[b], lh[b]);
}

__global__ void k_thr1(uint32_t* ws) {
    int r = threadIdx.x;
    if (r >= BATCH) return;
    const uint32_t* h = ws + OFF_H1 + r * BINS;
    uint32_t cum = 0, b1 = 0, r1 = KSEL;
    for (int b = BINS - 1; b >= 0; --b) {
        uint32_t c = h[b];
        if (cum + c >= (uint32_t)KSEL) { b1 = (uint32_t)b; r1 = KSEL - cum; break; }
        cum += c;
    }
    uint4* thr = (uint4*)(ws + OFF_THR);
    thr[r].x = b1;
    thr[r].y = r1;
}

__global__ void k_hist2(const float* __restrict__ pos,
                        const float* __restrict__ neg,
                        uint32_t* __restrict__ ws) {
    __shared__ uint32_t lh[BINS];
    for (int b = threadIdx.x; b < BINS; b += BLOCK) lh[b] = 0u;
    __syncthreads();

    const int row = blockIdx.x / ROWBLK;
    const int bir = blockIdx.x % ROWBLK;
    const uint32_t b1 = ((const uint4*)(ws + OFF_THR))[row].x;
    const float4* p4 = (const float4*)(pos + (size_t)row * NROW);
    const float4* n4 = (const float4*)(neg + (size_t)row * NROW);
    const int base = bir * 1024 + threadIdx.x;

    #pragma unroll
    for (int i = 0; i < 4; ++i) {
        int f = base + i * BLOCK;
        float4 P = p4[f];
        float4 Q = n4[f];
        float o; uint32_t bi;
        map_elem(P.x, Q.x, o, bi); if ((bi >> 21) == b1) atomicAdd(&lh[(bi >> 10) & 0x7FFu], 1u);
        map_elem(P.y, Q.y, o, bi); if ((bi >> 21) == b1) atomicAdd(&lh[(bi >> 10) & 0x7FFu], 1u);
        map_elem(P.z, Q.z, o, bi); if ((bi >> 21) == b1) atomicAdd(&lh[(bi >> 10) & 0x7FFu], 1u);
        map_elem(P.w, Q.w, o, bi); if ((bi >> 21) == b1) atomicAdd(&lh[(bi >> 10) & 0x7FFu], 1u);
    }
    __syncthreads();

    uint32_t* gh = ws + OFF_H2 + row * BINS;
    for (int b = threadIdx.x; b < BINS; b += BLOCK)
        if (lh[b]) atomicAdd(&gh[b], lh[b]);
}

__global__ void k_thr2(uint32_t* ws) {
    int r = threadIdx.x;
    if (r >= BATCH) return;
    uint4* thr = (uint4*)(ws + OFF_THR);
    const uint32_t target = thr[r].y;               // >= 1 by construction
    const uint32_t* h = ws + OFF_H2 + r * BINS;
    uint32_t cum = 0, b2 = 0, r2 = 0;
    for (int b = BINS - 1; b >= 0; --b) {
        uint32_t c = h[b];
        if (cum + c >= target) { b2 = (uint32_t)b; r2 = target - cum; break; }
        cum += c;
    }
    thr[r].z = b2;
    thr[r].w = r2;
}

__global__ void k_select(const float* __restrict__ pos,
                         const float* __restrict__ neg,
                         uint32_t* __restrict__ ws) {
    __shared__ float wsum[BLOCK / 32];

    const int row = blockIdx.x / ROWBLK;
    const int bir = blockIdx.x % ROWBLK;
    const uint4 tv = ((const uint4*)(ws + OFF_THR))[row];
    const uint32_t t22 = (tv.x << 11) | tv.z;       // 22-bit threshold key

    uint32_t* tcnt = ws + OFF_TCNT;
    uint32_t* tidx = ws + OFF_TIDX + row * TIECAP;
    float*    tval = (float*)(ws + OFF_TVAL) + row * TIECAP;

    const float4* p4 = (const float4*)(pos + (size_t)row * NROW);
    const float4* n4 = (const float4*)(neg + (size_t)row * NROW);
    const int base = bir * 1024 + threadIdx.x;

    float acc = 0.0f;
    #pragma unroll
    for (int i = 0; i < 4; ++i) {
        int f = base + i * BLOCK;
        float4 P = p4[f];
        float4 Q = n4[f];
        float po[4] = {P.x, P.y, P.z, P.w};
        float no[4] = {Q.x, Q.y, Q.z, Q.w};
        #pragma unroll
        for (int j = 0; j < 4; ++j) {
            float o; uint32_t bi;
            map_elem(po[j], no[j], o, bi);
            uint32_t e22 = bi >> 10;
            if (e22 > t22) {
                acc += o;                            // strictly above threshold
            } else if (e22 == t22) {                 // 22-bit tie: spill
                uint32_t slot = atomicAdd(&tcnt[row], 1u);
                if (slot < TIECAP) {
                    tidx[slot] = (uint32_t)(f * 4 + j);
                    tval[slot] = o;
                }
            }
        }
    }

    // EXEC is all-ones here (exact grid, unconditional path) -> WMMA legal
    float s32 = wave_sum_wmma(acc);
    if ((threadIdx.x & 31) == 0) wsum[threadIdx.x >> 5] = s32;
    __syncthreads();
    if (threadIdx.x == 0) {
        float s = 0.0f;
        #pragma unroll
        for (int w = 0; w < BLOCK / 32; ++w) s += wsum[w];
        atomicAdd((float*)(ws + OFF_RSUM) + row, s);
    }
}

__global__ void k_final(uint32_t* __restrict__ ws, float* __restrict__ out) {
    __shared__ float rs[BATCH];
    const int r = threadIdx.x;
    if (r < BATCH) {
        uint32_t cnt  = ws[OFF_TCNT + r];
        if (cnt > TIECAP) cnt = TIECAP;
        uint32_t need = ((const uint4*)(ws + OFF_THR))[r].w;
        const uint32_t* tidx = ws + OFF_TIDX + r * TIECAP;
        const float*    tval = (const float*)(ws + OFF_TVAL) + r * TIECAP;

        float s = ((const float*)(ws + OFF_RSUM))[r];
        // deterministically take the `need` ties with smallest flat index
        uint32_t last = 0; bool first = true;
        for (uint32_t i = 0; i < need; ++i) {
            uint32_t best = 0xFFFFFFFFu; float bv = 0.0f;
            for (uint32_t j = 0; j < cnt; ++j) {
                uint32_t ix = tidx[j];
                if ((first || ix > last) && ix < best) { best = ix; bv = tval[j]; }
            }
            if (best == 0xFFFFFFFFu) break;
            s += bv; last = best; first = false;
        }
        rs[r] = s;
    }
    __syncthreads();
    if (threadIdx.x == 0) {
        float t = 0.0f;
        for (int i = 0; i < BATCH; ++i) t += rs[i];
        out[0] = t / (float)(BATCH * KSEL);
    }
}

// ---------------- host launcher ----------------

extern "C" void kernel_launch(void* const* d_in, const int* in_sizes, int n_in,
                              void* d_out, int out_size, void* d_ws, size_t ws_size,
                              hipStream_t stream) {
    (void)in_sizes; (void)n_in; (void)out_size; (void)ws_size;
    const float* pos = (const float*)d_in[0];
    const float* neg = (const float*)d_in[1];
    float* out  = (float*)d_out;
    uint32_t* ws = (uint32_t*)d_ws;

    const int grid = BATCH * ROWBLK;   // 4096 blocks, one row per 256 blocks

    k_init  <<<(ZERO_WORDS + BLOCK - 1) / BLOCK, BLOCK, 0, stream>>>(ws);
    k_hist1 <<<grid, BLOCK, 0, stream>>>(pos, neg, ws);
    k_thr1  <<<1, 32, 0, stream>>>(ws);
    k_hist2 <<<grid, BLOCK, 0, stream>>>(pos, neg, ws);
    k_thr2  <<<1, 32, 0, stream>>>(ws);
    k_select<<<grid, BLOCK, 0, stream>>>(pos, neg, ws);
    k_final <<<1, 32, 0, stream>>>(ws, out);
}